// HybridLinear_24721831756588
// MI455X (gfx1250) — compile-verified
//
#include <hip/hip_runtime.h>

// Problem constants (from reference)
#define OUT_DIM 4096
#define IN_DIM  4096
#define RANK    256
#define BATCH   512      // 4 * 128 rows of x
#define NNZ     262144

// Block tiling: 256 threads = 8 waves (4 M x 2 N), block tile 128x128.
// Wave strip: MT x NT = 2 x 4 tiles of 16x16 (32 rows x 64 cols).
// K is consumed in BK=32 chunks staged through LDS.
#define MT 2
#define NT 4
#define BM 128
#define BN 128
#define BK 32
#define LDSW 36   // padded LDS row stride (floats): conflict-free + 16B aligned

typedef __attribute__((ext_vector_type(2))) float v2f;
typedef __attribute__((ext_vector_type(4))) float v4f;
typedef __attribute__((ext_vector_type(8))) float v8f;

// Workspace layout (floats):
//   [0,          16777216)  W    : OUT_DIM x IN_DIM, row-major (o*IN + i)
//   [16777216,  +1048576 )  U_s  : OUT_DIM x RANK,  row-major, pre-scaled by S
//   [17825792,  +1048576 )  VhT  : IN_DIM  x RANK,  row-major (Vh transposed)
// Requires ws_size >= 72 MiB.
#define W_OFF   0
#define US_OFF  (OUT_DIM * IN_DIM)
#define VHT_OFF (OUT_DIM * IN_DIM + OUT_DIM * RANK)

// ---------------------------------------------------------------------------
// 1) Codebook gather: U_s[o][r] = U_cb[U_idx[o*32 + r/8]][r%8] * S[r]
//    VhT[i][r]        = Vh_cb[Vh_idx[r*512 + i/8]][i%8]
//    Both factors are emitted K-contiguous for clean tile staging.
// ---------------------------------------------------------------------------
__global__ void gather_factors(const int* __restrict__ Ui,
                               const float* __restrict__ Ucb,
                               const float* __restrict__ S,
                               const int* __restrict__ Vi,
                               const float* __restrict__ Vcb,
                               float* __restrict__ Us,
                               float* __restrict__ VhT) {
    int t = blockIdx.x * blockDim.x + threadIdx.x;
    if (t < OUT_DIM * RANK) {
        int o = t >> 8;          // /RANK
        int r = t & 255;         // %RANK
        int sub = o * (RANK / 8) + (r >> 3);
        Us[t] = Ucb[Ui[sub] * 8 + (r & 7)] * S[r];
    } else {
        int t2 = t - OUT_DIM * RANK;    // flat over IN_DIM x RANK (i-major)
        int i = t2 >> 8;
        int r = t2 & 255;
        // original flat index in Vh (r-major over [RANK, IN]) = r*IN + i
        int sub = r * (IN_DIM / 8) + (i >> 3);
        VhT[t2] = Vcb[Vi[sub] * 8 + (i & 7)];
    }
}

// ---------------------------------------------------------------------------
// LDS-tiled block GEMM:  D[128x128 tile] = A[M][K] * Bt[N][K]^T
//   blockIdx.y -> M tile, blockIdx.x -> N tile.
//   Per K-chunk: stage A-panel (128xBK) and B-panel (128xBK) into LDS,
//   then each wave runs BK/4 = 8 WMMA K-steps from LDS fragments.
// A fragment layout (16x4 f32): lanes 0-15 -> M=lane,    K = k0+0/1
//                               lanes16-31 -> M=lane-16, K = k0+2/3
// C/D layout: VGPR v, lanes 0-15: M=v, N=lane; lanes 16-31: M=v+8, N=lane-16
// ---------------------------------------------------------------------------
__global__ __launch_bounds__(256)
void tile_gemm(const float* __restrict__ A,
               const float* __restrict__ Bt,
               float* __restrict__ D,
               int K, int ldD) {
    __shared__ float As[BM * LDSW];
    __shared__ float Bs[BN * LDSW];

    const int mBase = blockIdx.y * BM;
    const int nBase = blockIdx.x * BN;

    const int lane = threadIdx.x & 31;
    const int wave = threadIdx.x >> 5;      // 0..7
    const int m0 = (wave >> 1) * (MT * 16); // 0,32,64,96
    const int n0 = (wave & 1) * (NT * 16);  // 0,64

    // staging assignment: 2 threads per row, 16 contiguous floats each
    const int srow = threadIdx.x >> 1;          // 0..127
    const int scol = (threadIdx.x & 1) << 4;    // 0 or 16
    const float* agp = A  + (size_t)(mBase + srow) * K + scol;
    const float* bgp = Bt + (size_t)(nBase + srow) * K + scol;
    float* asp = &As[srow * LDSW + scol];
    float* bsp = &Bs[srow * LDSW + scol];

    // fragment read base offsets
    const int r0 = lane & 15;
    const int kb = (lane >> 4) << 1;

    v8f acc[MT][NT];
#pragma unroll
    for (int i = 0; i < MT; ++i)
#pragma unroll
        for (int j = 0; j < NT; ++j)
            acc[i][j] = (v8f){};

    for (int kc = 0; kc < K; kc += BK) {
        __syncthreads();   // previous chunk's readers done before overwrite
#pragma unroll
        for (int q = 0; q < 16; q += 4) {
            *(v4f*)(asp + q) = *(const v4f*)(agp + kc + q);
            *(v4f*)(bsp + q) = *(const v4f*)(bgp + kc + q);
        }
        __syncthreads();   // panels visible to all waves

#pragma unroll
        for (int kk = 0; kk < BK; kk += 4) {
            v2f a[MT], b[NT];
#pragma unroll
            for (int i = 0; i < MT; ++i)
                a[i] = *(const v2f*)&As[(m0 + i * 16 + r0) * LDSW + kk + kb];
#pragma unroll
            for (int j = 0; j < NT; ++j)
                b[j] = *(const v2f*)&Bs[(n0 + j * 16 + r0) * LDSW + kk + kb];
#pragma unroll
            for (int i = 0; i < MT; ++i)
#pragma unroll
                for (int j = 0; j < NT; ++j)
                    acc[i][j] = __builtin_amdgcn_wmma_f32_16x16x4_f32(
                        /*neg_a=*/false, a[i], /*neg_b=*/false, b[j],
                        /*c_mod=*/(short)0, acc[i][j],
                        /*reuse_a=*/false, /*reuse_b=*/false);
        }
    }

    const int rb = (lane >> 4) << 3;
#pragma unroll
    for (int i = 0; i < MT; ++i)
#pragma unroll
        for (int j = 0; j < NT; ++j) {
            float* dp = D + (size_t)(mBase + m0 + i * 16 + rb) * ldD
                          + (nBase + n0 + j * 16 + r0);
#pragma unroll
            for (int v = 0; v < 8; ++v)
                dp[(size_t)v * ldD] = acc[i][j][v];
        }
}

// ---------------------------------------------------------------------------
// 3) Sparse residual scatter-add into flat W (matches weight.reshape(-1))
// ---------------------------------------------------------------------------
__global__ void residual_scatter(const int* __restrict__ idx,
                                 const float* __restrict__ val,
                                 float* __restrict__ W) {
    int t = blockIdx.x * blockDim.x + threadIdx.x;
    if (t < NNZ) atomicAdd(&W[idx[t]], val[t]);
}

// ---------------------------------------------------------------------------
extern "C" void kernel_launch(void* const* d_in, const int* in_sizes, int n_in,
                              void* d_out, int out_size, void* d_ws, size_t ws_size,
                              hipStream_t stream) {
    const float* x    = (const float*)d_in[0];
    const int*   Ui   = (const int*)  d_in[1];
    const float* Ucb  = (const float*)d_in[2];
    const float* S    = (const float*)d_in[3];
    const int*   Vi   = (const int*)  d_in[4];
    const float* Vcb  = (const float*)d_in[5];
    const int*   ridx = (const int*)  d_in[6];
    const float* rval = (const float*)d_in[7];

    float* ws  = (float*)d_ws;
    float* W   = ws + W_OFF;
    float* Us  = ws + US_OFF;
    float* VhT = ws + VHT_OFF;
    float* out = (float*)d_out;

    // 1) gather both factor matrices (2 * 1M elements)
    {
        int total = OUT_DIM * RANK + IN_DIM * RANK;   // 2,097,152
        gather_factors<<<total / 256, 256, 0, stream>>>(Ui, Ucb, S, Vi, Vcb, Us, VhT);
    }
    // 2) W = U_s @ Vh : M=4096, N=4096, K=256 -> 32 x 32 tile grid
    tile_gemm<<<dim3(IN_DIM / BN, OUT_DIM / BM), 256, 0, stream>>>(
        Us, VhT, W, RANK, IN_DIM);
    // 3) sparse residual scatter-add (W is L2-resident: 64 MB < 192 MB)
    residual_scatter<<<NNZ / 256, 256, 0, stream>>>(ridx, rval, W);
    // 4) out = x @ W^T : M=512, N=4096, K=4096 -> 32 x 4 tile grid
    tile_gemm<<<dim3(OUT_DIM / BN, BATCH / BM), 256, 0, stream>>>(
        x, W, out, IN_DIM, OUT_DIM);
}